// label_assign_60120952210067
// MI455X (gfx1250) — compile-verified
//
#include <hip/hip_runtime.h>

// ---------------- problem constants ----------------
#define M_TGT       4096
#define NUM_CLS     20

// output layout (float offsets), concat of 6 tuple members in return order
#define OFF_CLSP    0           // (3*M, 20)  = 245760
#define OFF_REGP    245760      // (3*M, 4)   =  49152
#define OFF_OBJ     294912      // (1032192,1)
#define OFF_CLST    1327104     // (3*M, 20)
#define OFF_REGT    1572864     // (3*M, 4)
#define OFF_OBJT    1622016     // (1032192,1)

// per-level obj/obj_t bases within their 1032192-element blocks
__device__ __forceinline__ int level_objbase(int l) {
    return (l == 0) ? 0 : (l == 1) ? 49152 : 245760;
}

// =====================================================================
// Kernel A: per level, extract obj = reg[...,4] and zero obj_t.
// CDNA5 path: stage 1024 cells (5120 floats = 20 KB) of reg into LDS with
// async global->LDS b128 transfers (ASYNCcnt), then coalesced float4 stores.
// Launch: grid = nCells/1024 blocks of 256 threads.
// =====================================================================
__global__ __launch_bounds__(256) void obj_extract_zero_kernel(
    const float* __restrict__ reg,       // level reg tensor, 5 floats/cell
    float* __restrict__ obj_out,         // level slice of obj output
    float* __restrict__ objt_out)        // level slice of obj_t output
{
    __shared__ float smem[5120];         // 1024 cells * 5 floats = 20480 B

    const unsigned t        = threadIdx.x;
    const unsigned tileBase = blockIdx.x * 1024u;            // first cell of tile
    const float*   src      = reg + (size_t)tileBase * 5u;   // 16B aligned (tile*20B)

    const unsigned ldsBase = (unsigned)(uintptr_t)(void*)smem; // low 32 bits = LDS offset
    unsigned ldsAddr = ldsBase + t * 16u;
    unsigned gOff    = t * 16u;

    // 1280 lane-transfers of 16B each = 20480 B; 5 async b128 per thread.
#pragma unroll
    for (int k = 0; k < 5; ++k) {
        asm volatile("global_load_async_to_lds_b128 %0, %1, %2"
                     :: "v"(ldsAddr), "v"(gOff), "s"(src)
                     : "memory");
        ldsAddr += 4096u;
        gOff    += 4096u;
    }
    asm volatile("s_wait_asynccnt 0" ::: "memory");   // this wave's async writes done
    __syncthreads();                                  // all waves' LDS data visible

    // thread t owns cells tileBase+4t .. +3 ; obj lives at lds dword 5*c+4
    float4 o;
    o.x = smem[20u * t + 4u];
    o.y = smem[20u * t + 9u];
    o.z = smem[20u * t + 14u];
    o.w = smem[20u * t + 19u];

    reinterpret_cast<float4*>(obj_out  + tileBase)[t] = o;
    reinterpret_cast<float4*>(objt_out + tileBase)[t] = make_float4(0.f, 0.f, 0.f, 0.f);
}

// =====================================================================
// Kernel B: one thread per (level, target). 12288 threads.
// IoU vs 3 anchors -> argmax/mask, gather cls/reg rows, encode deltas,
// one-hot class target, scatter mask into obj_t (after kernel A zeroed it).
// =====================================================================
__global__ __launch_bounds__(256) void assign_targets_kernel(
    const float* __restrict__ targets,   // (4096, 7)
    const float* __restrict__ anchors,   // (3, 3, 4)
    const float* __restrict__ cls0, const float* __restrict__ reg0,
    const float* __restrict__ cls1, const float* __restrict__ reg1,
    const float* __restrict__ cls2, const float* __restrict__ reg2,
    float* __restrict__ out)
{
    const int t = blockIdx.x * blockDim.x + threadIdx.x;   // 0..12287
    const int l = t >> 12;          // level (uniform per 256-thread block)
    const int j = t & (M_TGT - 1);  // target index

    const int   S      = 32 << l;                 // 32, 64, 128
    const float stride = 32.0f / (float)(1 << l); // 32, 16, 8
    const float inv    = 1.0f / stride;

    const float* cls = (l == 0) ? cls0 : (l == 1) ? cls1 : cls2;
    const float* reg = (l == 0) ? reg0 : (l == 1) ? reg1 : reg2;

    // ---- target row ----
    const float* tg = targets + j * 7;
    const float x0 = tg[0], y0 = tg[1], x1 = tg[2], y1 = tg[3];
    const int   ci  = (int)tg[4];
    const int   bid = (int)tg[6];

    const float w = x1 - x0, h = y1 - y0;
    const float bx0 = -0.5f * w * inv, by0 = -0.5f * h * inv;
    const float bx1 =  0.5f * w * inv, by1 =  0.5f * h * inv;
    const float cx  = 0.5f * (x0 + x1) * inv;
    const float cy  = 0.5f * (y0 + y1) * inv;
    const int   gx = (int)cx, gy = (int)cy;
    const float areaB = (bx1 - bx0) * (by1 - by0);

    // ---- IoU vs 3 anchors, first-max argmax ----
    const float* an = anchors + l * 12;
    float best = -1.0f;
    int   arg  = 0;
#pragma unroll
    for (int a = 0; a < 3; ++a) {
        const float ax0 = an[a * 4 + 0], ay0 = an[a * 4 + 1];
        const float ax1 = an[a * 4 + 2], ay1 = an[a * 4 + 3];
        const float iw = fmaxf(fminf(ax1, bx1) - fmaxf(ax0, bx0), 0.0f);
        const float ih = fmaxf(fminf(ay1, by1) - fmaxf(ay0, by0), 0.0f);
        const float inter = iw * ih;
        const float areaA = (ax1 - ax0) * (ay1 - ay0);
        const float iou = inter / (areaA + areaB - inter + 1e-9f);
        if (iou > best) { best = iou; arg = a; }
    }
    const float mask = (best > 0.3f) ? 1.0f : 0.0f;

    const long cell = ((long)((bid * 3 + arg) * S + gx)) * S + gy;
    const float* clsRow = cls + cell * NUM_CLS;   // 80B-aligned
    const float* regRow = reg + cell * 5;
    __builtin_prefetch(clsRow, 0, 0);             // global_prefetch_b8
    __builtin_prefetch(regRow, 0, 0);

    const long row  = (long)l * M_TGT + j;
    float* clsP = out + OFF_CLSP + row * NUM_CLS;
    float* regP = out + OFF_REGP + row * 4;
    float* clsT = out + OFF_CLST + row * NUM_CLS;
    float* regT = out + OFF_REGT + row * 4;

    // ---- cls_p: gathered 20-float row * mask ----
    const float4* cs = reinterpret_cast<const float4*>(clsRow);
    float4*       cp = reinterpret_cast<float4*>(clsP);
#pragma unroll
    for (int k = 0; k < 5; ++k) {
        float4 v = cs[k];
        v.x *= mask; v.y *= mask; v.z *= mask; v.w *= mask;
        cp[k] = v;
    }

    // ---- reg_p ----
    float4 rp;
    rp.x = regRow[0] * mask; rp.y = regRow[1] * mask;
    rp.z = regRow[2] * mask; rp.w = regRow[3] * mask;
    *reinterpret_cast<float4*>(regP) = rp;

    // ---- reg_t = encode(anchor[arg], target/stride) * mask ----
    const float ax0 = an[arg * 4 + 0], ay0 = an[arg * 4 + 1];
    const float ax1 = an[arg * 4 + 2], ay1 = an[arg * 4 + 3];
    const float ew = fmaxf(ax1 - ax0, 1.0f);
    const float eh = fmaxf(ay1 - ay0, 1.0f);
    const float gtx0 = x0 * inv, gty0 = y0 * inv;
    const float gtx1 = x1 * inv, gty1 = y1 * inv;
    const float gw = fmaxf(gtx1 - gtx0, 1.0f);
    const float gh = fmaxf(gty1 - gty0, 1.0f);
    const float gcx = gtx0 + 0.5f * gw;
    const float gcy = gty0 + 0.5f * gh;
    const float dx = gcx - (float)(short)gcx;     // int16 truncation quirk
    const float dy = gcy - (float)(short)gcy;
    float4 rt;
    rt.x = dx * mask;
    rt.y = dy * mask;
    rt.z = logf(gw / ew) * mask;
    rt.w = logf(gh / eh) * mask;
    *reinterpret_cast<float4*>(regT) = rt;

    // ---- cls_t: one-hot * mask ----
    float4* ct = reinterpret_cast<float4*>(clsT);
#pragma unroll
    for (int k = 0; k < 5; ++k) {
        float4 v;
        v.x = (ci == 4 * k + 0) ? mask : 0.0f;
        v.y = (ci == 4 * k + 1) ? mask : 0.0f;
        v.z = (ci == 4 * k + 2) ? mask : 0.0f;
        v.w = (ci == 4 * k + 3) ? mask : 0.0f;
        ct[k] = v;
    }

    // ---- obj_t scatter (region pre-zeroed by kernel A) ----
    out[(long)OFF_OBJT + level_objbase(l) + cell] = mask;
}

// =====================================================================
extern "C" void kernel_launch(void* const* d_in, const int* in_sizes, int n_in,
                              void* d_out, int out_size, void* d_ws, size_t ws_size,
                              hipStream_t stream) {
    const float* targets = (const float*)d_in[0];
    const float* anchors = (const float*)d_in[1];
    const float* cls0    = (const float*)d_in[2];
    const float* reg0    = (const float*)d_in[3];
    const float* cls1    = (const float*)d_in[4];
    const float* reg1    = (const float*)d_in[5];
    const float* cls2    = (const float*)d_in[6];
    const float* reg2    = (const float*)d_in[7];
    float* out = (float*)d_out;

    float* obj  = out + OFF_OBJ;
    float* objt = out + OFF_OBJT;

    // Kernel A: obj extraction + obj_t zeroing (async global->LDS staging).
    obj_extract_zero_kernel<<<49152  / 1024, 256, 0, stream>>>(reg0, obj + 0,      objt + 0);
    obj_extract_zero_kernel<<<196608 / 1024, 256, 0, stream>>>(reg1, obj + 49152,  objt + 49152);
    obj_extract_zero_kernel<<<786432 / 1024, 256, 0, stream>>>(reg2, obj + 245760, objt + 245760);

    // Kernel B: per-target assignment (runs after A on the stream, so the
    // obj_t scatter lands on a zeroed buffer).
    assign_targets_kernel<<<(3 * M_TGT) / 256, 256, 0, stream>>>(
        targets, anchors, cls0, reg0, cls1, reg1, cls2, reg2, out);
}